// AuxLoss_54443005444679
// MI455X (gfx1250) — compile-verified
//
#include <hip/hip_runtime.h>
#include <hip/hip_bf16.h>
#include <math.h>

// Problem constants (from reference setup): y_pred [B, V] f32, target [B, 2] i32, weight [2] f32.
#define VOCAB          4096
#define TOPK           2
#define ROWS_PER_BLOCK 8          // 8 waves per block, one wave per row
#define THREADS        256        // 8 wave32 waves
#define LBOUND         (-100.0f)

typedef float v8f __attribute__((ext_vector_type(8)));
typedef float v2f __attribute__((ext_vector_type(2)));

// ---------------------------------------------------------------------------
// Kernel 1: per-row loss. Stage one 16KB row per wave into LDS with CDNA5
// async Global->LDS DMA (ASYNCcnt), then two passes (max, exp-sum) from LDS.
// Dynamic LDS: ROWS_PER_BLOCK * VOCAB * 4 = 128 KB -> 2 blocks per 320KB WGP.
// ---------------------------------------------------------------------------
__global__ __launch_bounds__(THREADS) void row_loss_kernel(
    const float* __restrict__ y,
    const int*   __restrict__ tgt,
    const float* __restrict__ wt,
    float*       __restrict__ per_row,
    int B)
{
    extern __shared__ float lds[];
    const int tid  = threadIdx.x;
    const int wave = tid >> 5;
    const int lane = tid & 31;
    const int row  = blockIdx.x * ROWS_PER_BLOCK + wave;
    if (row >= B) return;                       // wave-uniform exit

    float* myrow = &lds[wave * VOCAB];

    // ---- async DMA: 16KB row -> LDS. 32 wave-instructions x (32 lanes x 16B).
    // GVS addressing: mem = SGPR64(y) + VGPR32(goff); LDS addr from VDST VGPR.
    uint32_t ldsa = (uint32_t)(size_t)myrow + (uint32_t)(lane * 16);
    uint32_t goff = (uint32_t)row * (uint32_t)(VOCAB * 4) + (uint32_t)(lane * 16);
#pragma unroll
    for (int i = 0; i < 32; ++i) {
        asm volatile("global_load_async_to_lds_b128 %0, %1, %2"
                     :: "v"(ldsa), "v"(goff), "s"(y)
                     : "memory");
        ldsa += 512u;
        goff += 512u;
    }
    asm volatile("s_wait_asynccnt 0x0" ::: "memory");

    // ---- pass 1: row max (128 elems/lane via ds_load_b128)
    const float4* rv = (const float4*)myrow;
    float m = -INFINITY;
#pragma unroll 4
    for (int i = 0; i < 32; ++i) {
        float4 v = rv[lane + 32 * i];
        m = fmaxf(m, fmaxf(fmaxf(v.x, v.y), fmaxf(v.z, v.w)));
    }
#pragma unroll
    for (int off = 16; off > 0; off >>= 1)
        m = fmaxf(m, __shfl_xor(m, off, 32));

    // ---- pass 2: Z = sum exp(x - m)
    float s0 = 0.f, s1 = 0.f, s2 = 0.f, s3 = 0.f;
#pragma unroll 4
    for (int i = 0; i < 32; ++i) {
        float4 v = rv[lane + 32 * i];
        s0 += __expf(v.x - m);
        s1 += __expf(v.y - m);
        s2 += __expf(v.z - m);
        s3 += __expf(v.w - m);
    }
    float z = (s0 + s1) + (s2 + s3);
#pragma unroll
    for (int off = 16; off > 0; off >>= 1)
        z += __shfl_xor(z, off, 32);

    // ---- gather targets from the staged LDS row, compute per-sample loss
    if (lane == 0) {
        const int   t0 = tgt[row * TOPK + 0];
        const int   t1 = tgt[row * TOPK + 1];
        const float w0 = wt[0];
        const float w1 = wt[1];
        const bool  m0 = (t0 != -1);
        const bool  m1 = (t1 != -1);
        float acc = 0.0f;
        if (m0) acc += w0 * __expf(myrow[t0] - m);
        if (m1) acc += w1 * __expf(myrow[t1] - m);
        const float dot = acc / z;
        const int   tl  = (int)m0 + (int)m1;
        // suffix[j] = sum(weight[j:]); discount = 1 - suffix[TOPK - tl] for tl < TOPK
        const float discount = (tl == TOPK) ? 1.0f : (tl == 1 ? (1.0f - w1) : 1.0f);
        const float v = __logf(dot / discount);            // log(0) -> -inf, clamped below
        per_row[row] = -fmaxf(v, LBOUND);
    }
}

// ---------------------------------------------------------------------------
// Kernel 2: deterministic 16384 -> 1 reduction on the matrix unit.
// V_WMMA_F32_16X16X4_F32 with B = ones: D[m,n] = sum_k A[m,k] + C[m,n].
// Each WMMA folds 64 f32 values (2 per lane) into the f32 accumulator; fixed
// order => bitwise-deterministic across graph replays (no float atomics).
// ---------------------------------------------------------------------------
__global__ __launch_bounds__(THREADS) void wmma_reduce_kernel(
    const float* __restrict__ per_row,
    float*       __restrict__ out,
    int n, float scale)
{
    __shared__ float wsum[THREADS / 32];
    const int tid  = threadIdx.x;
    const int wave = tid >> 5;
    const int lane = tid & 31;
    const int chunk = (n + (THREADS / 32) - 1) / (THREADS / 32);
    const int base  = wave * chunk;

    v8f c    = {0.f, 0.f, 0.f, 0.f, 0.f, 0.f, 0.f, 0.f};
    v2f ones = {1.0f, 1.0f};

    for (int i = 0; i < chunk; i += 64) {
        int i0 = base + i + lane;
        int i1 = i0 + 32;
        // branch-free bounds handling (EXEC must stay all-ones for WMMA)
        int   j0 = (i0 < n) ? i0 : 0;
        int   j1 = (i1 < n) ? i1 : 0;
        float x0 = per_row[j0];
        float x1 = per_row[j1];
        v2f a;
        a.x = (i0 < n) ? x0 : 0.0f;
        a.y = (i1 < n) ? x1 : 0.0f;
        // (neg_a, A, neg_b, B, c_mod, C, reuse_a, reuse_b)
        c = __builtin_amdgcn_wmma_f32_16x16x4_f32(false, a, false, ones,
                                                  (short)0, c, false, false);
    }

    // Column n=0 of C: rows 0..7 live in VGPRs 0..7 of lane 0, rows 8..15 in lane 16.
    float s   = ((c[0] + c[1]) + (c[2] + c[3])) + ((c[4] + c[5]) + (c[6] + c[7]));
    float tot = __shfl(s, 0, 32) + __shfl(s, 16, 32);
    if (lane == 0) wsum[wave] = tot;
    __syncthreads();
    if (tid == 0) {
        float t = 0.0f;
        for (int i = 0; i < THREADS / 32; ++i) t += wsum[i];
        out[0] = t * scale;
    }
}

// ---------------------------------------------------------------------------
extern "C" void kernel_launch(void* const* d_in, const int* in_sizes, int n_in,
                              void* d_out, int out_size, void* d_ws, size_t ws_size,
                              hipStream_t stream)
{
    const float* y   = (const float*)d_in[0];   // [B, V] f32
    const int*   tgt = (const int*)d_in[1];     // [B, 2] i32
    const float* wt  = (const float*)d_in[2];   // [2] f32
    float* out       = (float*)d_out;           // [1] f32
    float* per_row   = (float*)d_ws;            // B floats of scratch

    const int B = in_sizes[0] / VOCAB;
    const int nblocks = (B + ROWS_PER_BLOCK - 1) / ROWS_PER_BLOCK;
    const size_t lds_bytes = (size_t)ROWS_PER_BLOCK * VOCAB * sizeof(float); // 128 KB

    row_loss_kernel<<<nblocks, THREADS, lds_bytes, stream>>>(y, tgt, wt, per_row, B);
    wmma_reduce_kernel<<<1, THREADS, 0, stream>>>(per_row, out, B, 1.0f / (float)B);
}